// MultiBoxLoss_10591389352247
// MI455X (gfx1250) — compile-verified
//
#include <hip/hip_runtime.h>
#include <math.h>

#define BATCH   32
#define NPRIOR  24564
#define NCLS    21
#define NOBJ    50
#define THRESHV 0.5f
#define NEG_RATIO 3
#define VAR0 0.1f
#define VAR1 0.2f
#define NTOT (BATCH * NPRIOR)
#define PREDW (4 + NCLS)     // 25 floats per (b,p)

#define NB2 1024             // blocks for streaming loss kernel
#define TB2 256

typedef __attribute__((ext_vector_type(2))) float v2f;
typedef __attribute__((ext_vector_type(8))) float v8f;

// ---------------------------------------------------------------------------
// Exact fp32 wave(32) sum on the matrix pipe:
//   A[m][k]: lane m (0..15) vgpr0 -> K=0, lanes 16..31 vgpr0 -> K=2 ; vgpr1 = 0
//   B = ones  =>  D[m][n] = x[m] + x[m+16] for every n.
//   D layout: lane L holds rows {r} (L<16) / {8+r} (L>=16) => sum 8 accs,
//   then one shfl_xor(16) completes the 16-row sum. Exact fp32 accumulate.
// ---------------------------------------------------------------------------
__device__ __forceinline__ float wave_sum32(float x) {
#if __has_builtin(__builtin_amdgcn_wmma_f32_16x16x4_f32)
  v2f a;  a[0] = x;   a[1] = 0.0f;
  v2f on; on[0] = 1.0f; on[1] = 1.0f;
  v8f c = {};
  c = __builtin_amdgcn_wmma_f32_16x16x4_f32(false, a, false, on, (short)0, c,
                                            false, false);
  float s = ((c[0] + c[1]) + (c[2] + c[3])) + ((c[4] + c[5]) + (c[6] + c[7]));
  s += __shfl_xor(s, 16, 32);
  return s;
#else
  for (int d = 16; d > 0; d >>= 1) x += __shfl_xor(x, d, 32);
  return x;
#endif
}

__device__ __forceinline__ int wave_sum32_i(int x) {
  for (int d = 16; d > 0; d >>= 1) x += __shfl_xor(x, d, 32);
  return x;
}

// ---------------------------------------------------------------------------
// Kernel 1: per-batch matching. One block per batch image.
//   - per-prior best truth (argmax over O, first-max tie-break)
//   - per-truth best prior via LDS ds_max_u64 on (iou_bits<<32 | ~p)
//   - force-match override (serialized in j order => deterministic)
//   - num_pos per batch
// ---------------------------------------------------------------------------
__global__ __launch_bounds__(1024) void k_match(
    const float* __restrict__ gt_boxes, const int* __restrict__ gt_labels,
    const float* __restrict__ priors, int* __restrict__ conf_t,
    int* __restrict__ midx, int* __restrict__ numpos) {
  const int b = blockIdx.x;
  __shared__ float tx1[NOBJ], ty1[NOBJ], tx2[NOBJ], ty2[NOBJ], areaA[NOBJ];
  __shared__ int lab[NOBJ];
  __shared__ unsigned long long bestp[NOBJ];
  __shared__ int cntLds[32];

  for (int j = threadIdx.x; j < NOBJ; j += blockDim.x) {
    const float4 t = ((const float4*)gt_boxes)[(size_t)b * NOBJ + j];
    tx1[j] = t.x - 0.5f * t.z;  ty1[j] = t.y - 0.5f * t.w;
    tx2[j] = t.x + 0.5f * t.z;  ty2[j] = t.y + 0.5f * t.w;
    areaA[j] = t.z * t.w;
    lab[j] = gt_labels[b * NOBJ + j];
    bestp[j] = 0ull;
  }
  __syncthreads();

  for (int p = threadIdx.x; p < NPRIOR; p += blockDim.x) {
    const float4 q = ((const float4*)priors)[p];
    const float px1 = q.x - 0.5f * q.z, py1 = q.y - 0.5f * q.w;
    const float px2 = q.x + 0.5f * q.z, py2 = q.y + 0.5f * q.w;
    const float areaB = q.z * q.w;
    float best_ov = -1.0f; int best_j = 0;
    const unsigned lowkey = ~(unsigned)p;   // smaller p => larger key (first-max)
    for (int j = 0; j < NOBJ; ++j) {
      const float ix1 = fmaxf(tx1[j], px1), iy1 = fmaxf(ty1[j], py1);
      const float ix2 = fminf(tx2[j], px2), iy2 = fminf(ty2[j], py2);
      const float iw = fmaxf(ix2 - ix1, 0.0f), ih = fmaxf(iy2 - iy1, 0.0f);
      const float inter = iw * ih;
      const float iou = inter / (areaA[j] + areaB - inter);
      if (iou > best_ov) { best_ov = iou; best_j = j; }
      const unsigned long long key =
          ((unsigned long long)__float_as_uint(iou) << 32) | (unsigned long long)lowkey;
      if (key > bestp[j]) atomicMax(&bestp[j], key);   // monotone, order-free
    }
    conf_t[(size_t)b * NPRIOR + p] = (best_ov < THRESHV) ? 0 : lab[best_j];
    midx[(size_t)b * NPRIOR + p] = best_j;
  }
  __syncthreads();

  if (threadIdx.x == 0) {                 // sequential scatter: deterministic
    for (int j = 0; j < NOBJ; ++j) {
      const unsigned pstar = ~(unsigned)(bestp[j] & 0xffffffffull);
      conf_t[(size_t)b * NPRIOR + pstar] = lab[j];
      midx[(size_t)b * NPRIOR + pstar] = j;
    }
  }
  __syncthreads();

  int cnt = 0;
  for (int p = threadIdx.x; p < NPRIOR; p += blockDim.x)
    cnt += (conf_t[(size_t)b * NPRIOR + p] > 0) ? 1 : 0;
  cnt = wave_sum32_i(cnt);
  const int lane = threadIdx.x & 31, wid = threadIdx.x >> 5;
  if (lane == 0) cntLds[wid] = cnt;
  __syncthreads();
  if (threadIdx.x == 0) {
    int t = 0;
    for (int i = 0; i < (int)(blockDim.x >> 5); ++i) t += cntLds[i];
    numpos[b] = t;
  }
}

// ---------------------------------------------------------------------------
// Kernel 2: HBM-bound streaming pass over all 78.6 MB of predictions.
//   Per (b,p): logsumexp(21), CE, smooth-L1 if positive; mine[] for mining.
//   Per-block partial sums (fixed-order, WMMA wave reduce) -> deterministic.
// ---------------------------------------------------------------------------
__global__ __launch_bounds__(TB2) void k_loss(
    const float* __restrict__ pred, const float* __restrict__ gt_boxes,
    const float* __restrict__ priors, const int* __restrict__ conf_t,
    const int* __restrict__ midx, float* __restrict__ mine,
    float* __restrict__ part2) {
  __shared__ float red[16];
  float accL = 0.0f, accC = 0.0f;
  const int stride = NB2 * TB2;
  for (int i = blockIdx.x * TB2 + threadIdx.x; i < NTOT; i += stride) {
    const int b = i / NPRIOR;
    const int p = i - b * NPRIOR;
    const float* e = pred + (size_t)i * PREDW;
    __builtin_prefetch(e + (size_t)stride * PREDW, 0, 1);  // global_prefetch_b8

    float v[NCLS];
#pragma unroll
    for (int k = 0; k < NCLS; ++k) v[k] = e[4 + k];
    float m = v[0];
#pragma unroll
    for (int k = 1; k < NCLS; ++k) m = fmaxf(m, v[k]);
    float se = 0.0f;
#pragma unroll
    for (int k = 0; k < NCLS; ++k) se += __expf(v[k] - m);
    const float lse = m + __logf(se);

    const int c = conf_t[i];
    const float picked = e[4 + c];         // re-load (avoids dyn reg indexing)
    const float ce = lse - picked;

    if (c > 0) {
      accC += ce;
      mine[i] = 0.0f;
      const int j = midx[i];
      const float4 t = ((const float4*)gt_boxes)[(size_t)b * NOBJ + j];
      const float4 q = ((const float4*)priors)[p];
      const float g0 = (t.x - q.x) / (VAR0 * q.z);
      const float g1 = (t.y - q.y) / (VAR0 * q.w);
      const float g2 = __logf(t.z / q.z) / VAR1;
      const float g3 = __logf(t.w / q.w) / VAR1;
      float s = 0.0f, d;
      d = fabsf(e[0] - g0); s += (d < 1.0f) ? 0.5f * d * d : (d - 0.5f);
      d = fabsf(e[1] - g1); s += (d < 1.0f) ? 0.5f * d * d : (d - 0.5f);
      d = fabsf(e[2] - g2); s += (d < 1.0f) ? 0.5f * d * d : (d - 0.5f);
      d = fabsf(e[3] - g3); s += (d < 1.0f) ? 0.5f * d * d : (d - 0.5f);
      accL += s;
    } else {
      mine[i] = fmaxf(ce, 0.0f);   // keep bits non-negative for radix select
    }
  }
  const float wl = wave_sum32(accL);
  const float wc = wave_sum32(accC);
  const int lane = threadIdx.x & 31, wid = threadIdx.x >> 5;
  if (lane == 0) { red[wid] = wl; red[8 + wid] = wc; }
  __syncthreads();
  if (threadIdx.x == 0) {
    float tl = 0.0f, tc = 0.0f;
    for (int i = 0; i < TB2 / 32; ++i) { tl += red[i]; tc += red[8 + i]; }
    part2[blockIdx.x * 2 + 0] = tl;
    part2[blockIdx.x * 2 + 1] = tc;
  }
}

// ---------------------------------------------------------------------------
// Kernel 3: per-batch top-K sum via 4-pass MSB radix select over float bits
// (all mine >= 0 so uint order == float order). Replaces both argsorts.
// ---------------------------------------------------------------------------
__global__ __launch_bounds__(256) void k_topk(
    const float* __restrict__ mine, const int* __restrict__ numpos,
    float* __restrict__ negsum) {
  const int b = blockIdx.x;
  __shared__ unsigned hist[256];
  __shared__ unsigned s_prefix;
  __shared__ int s_remain;
  __shared__ float rLds[8];
  const float* mb = mine + (size_t)b * NPRIOR;

  int K = NEG_RATIO * numpos[b];
  if (K > NPRIOR - 1) K = NPRIOR - 1;
  if (threadIdx.x == 0) { s_prefix = 0u; s_remain = K; }
  __syncthreads();
  if (K <= 0) { if (threadIdx.x == 0) negsum[b] = 0.0f; return; }

  for (int shift = 24; shift >= 0; shift -= 8) {
    hist[threadIdx.x] = 0u;            // blockDim == 256
    __syncthreads();
    const unsigned pref = s_prefix;
    const unsigned maskHi = (shift == 24) ? 0u : (0xFFFFFFFFu << (shift + 8));
    for (int p = threadIdx.x; p < NPRIOR; p += 256) {
      const unsigned vb = __float_as_uint(mb[p]);
      if ((vb & maskHi) == pref) atomicAdd(&hist[(vb >> shift) & 255u], 1u);
    }
    __syncthreads();
    if (threadIdx.x == 0) {
      int rem = s_remain, acc = 0; unsigned pick = 0u;
      for (int bkt = 255; bkt >= 0; --bkt) {
        const int ch = (int)hist[bkt];
        if (acc + ch >= rem) { pick = (unsigned)bkt; s_remain = rem - acc; break; }
        acc += ch;
      }
      s_prefix = pref | (pick << shift);
    }
    __syncthreads();
  }

  const unsigned T = s_prefix;            // K-th largest value (bits)
  const float Tval = __uint_as_float(T);
  float s = 0.0f;
  for (int p = threadIdx.x; p < NPRIOR; p += 256) {
    const float x = mb[p];
    if (__float_as_uint(x) > T) s += x;   // strictly-greater part
  }
  const float wsum = wave_sum32(s);
  const int lane = threadIdx.x & 31, wid = threadIdx.x >> 5;
  if (lane == 0) rLds[wid] = wsum;
  __syncthreads();
  if (threadIdx.x == 0) {
    float tot = 0.0f;
    for (int i = 0; i < 8; ++i) tot += rLds[i];
    negsum[b] = tot + (float)s_remain * Tval;   // ties at threshold
  }
}

// ---------------------------------------------------------------------------
// Kernel 4: deterministic finalize.
//   loss_c = sum_pos(ce) + sum(topK(mine))   (exact decomposition: mine==0 on pos)
// ---------------------------------------------------------------------------
__global__ __launch_bounds__(256) void k_final(
    const float* __restrict__ part2, const float* __restrict__ negsum,
    const int* __restrict__ numpos, float* __restrict__ out) {
  __shared__ float rl[8], rc[8];
  __shared__ int rn[8];
  float al = 0.0f, ac = 0.0f;
  int np = 0;
  for (int i = threadIdx.x; i < NB2; i += 256) {
    al += part2[i * 2 + 0];
    ac += part2[i * 2 + 1];
  }
  for (int i = threadIdx.x; i < BATCH; i += 256) {
    ac += negsum[i];
    np += numpos[i];
  }
  const float wl = wave_sum32(al);
  const float wc = wave_sum32(ac);
  np = wave_sum32_i(np);
  const int lane = threadIdx.x & 31, wid = threadIdx.x >> 5;
  if (lane == 0) { rl[wid] = wl; rc[wid] = wc; rn[wid] = np; }
  __syncthreads();
  if (threadIdx.x == 0) {
    float tl = 0.0f, tc = 0.0f; int tn = 0;
    for (int i = 0; i < 8; ++i) { tl += rl[i]; tc += rc[i]; tn += rn[i]; }
    const float bp = (float)tn;
    out[0] = tl / bp;
    out[1] = tc / bp;
  }
}

// ---------------------------------------------------------------------------
extern "C" void kernel_launch(void* const* d_in, const int* in_sizes, int n_in,
                              void* d_out, int out_size, void* d_ws,
                              size_t ws_size, hipStream_t stream) {
  (void)in_sizes; (void)n_in; (void)out_size; (void)ws_size;
  const float* pred   = (const float*)d_in[0];   // (B,P,25)
  const float* gtb    = (const float*)d_in[1];   // (B,O,4)
  const float* priors = (const float*)d_in[2];   // (P,4)
  const int*   gtl    = (const int*)d_in[3];     // (B,O)

  char* ws = (char*)d_ws;
  size_t off = 0;
  auto carve = [&](size_t bytes) -> void* {
    void* p = ws + off;
    off += (bytes + 255) & ~(size_t)255;
    return p;
  };
  int*   conf_t = (int*)  carve((size_t)NTOT * sizeof(int));
  int*   midx   = (int*)  carve((size_t)NTOT * sizeof(int));
  float* mine   = (float*)carve((size_t)NTOT * sizeof(float));
  int*   numpos = (int*)  carve(BATCH * sizeof(int));
  float* negsum = (float*)carve(BATCH * sizeof(float));
  float* part2  = (float*)carve((size_t)NB2 * 2 * sizeof(float));

  hipLaunchKernelGGL(k_match, dim3(BATCH), dim3(1024), 0, stream,
                     gtb, gtl, priors, conf_t, midx, numpos);
  hipLaunchKernelGGL(k_loss, dim3(NB2), dim3(TB2), 0, stream,
                     pred, gtb, priors, conf_t, midx, mine, part2);
  hipLaunchKernelGGL(k_topk, dim3(BATCH), dim3(256), 0, stream,
                     mine, numpos, negsum);
  hipLaunchKernelGGL(k_final, dim3(1), dim3(256), 0, stream,
                     part2, negsum, numpos, (float*)d_out);
}